// MambaBlock_9234179686552
// MI455X (gfx1250) — compile-verified
//
#include <hip/hip_runtime.h>
#include <hip/hip_bf16.h>

typedef __attribute__((ext_vector_type(2))) float v2f;
typedef __attribute__((ext_vector_type(8))) float v8f;

#define D_MODEL 1024
#define D_STATE 16
#define D_CONV  4
#define D_INNER 2048
#define BATCH   2
#define SEQ     1024
#define NROWS   (BATCH * SEQ)          // 2048 rows (b,s)
#define EPS_F   1.1920929e-7f

// ---------------------------------------------------------------------------
// CDNA5 async global->LDS copy support (ASYNCcnt-tracked, bypasses VGPRs).
// Builtin signature (from hipcc diagnostic): params are pointers to a 16-byte
// int vector: (addrspace(1) v4i*, addrspace(3) v4i*, imm offset, imm cpol).
// ---------------------------------------------------------------------------
#if __has_builtin(__builtin_amdgcn_global_load_async_to_lds_b128)
#define HAVE_ASYNC_LDS 1
typedef int v4i __attribute__((vector_size(16)));
typedef __attribute__((address_space(1))) v4i* gv4_ptr;   // pointer to global v4i
typedef __attribute__((address_space(3))) v4i* lv4_ptr;   // pointer to LDS v4i
__device__ __forceinline__ void async_cp16(const float* g, float* l) {
    __builtin_amdgcn_global_load_async_to_lds_b128((gv4_ptr)(void*)g, (lv4_ptr)(void*)l, 0, 0);
}
__device__ __forceinline__ void wait_async0() {
#if __has_builtin(__builtin_amdgcn_s_wait_asynccnt)
    __builtin_amdgcn_s_wait_asynccnt(0);
#else
    asm volatile("s_wait_asynccnt 0" ::: "memory");
#endif
}
#else
#define HAVE_ASYNC_LDS 0
#endif

// ---------------------------------------------------------------------------
// Kernel 1: RMSNorm: xn[row][k] = x[row][k] * rsqrt(mean(x^2)+eps) * norm_w[k]
// ---------------------------------------------------------------------------
__global__ __launch_bounds__(256)
void rmsnorm_kernel(const float* __restrict__ x, const float* __restrict__ normw,
                    float* __restrict__ xn) {
    __shared__ float red[256];
    const int row = blockIdx.x;
    const float* xr = x + (size_t)row * D_MODEL;
    float s = 0.f;
    for (int i = threadIdx.x; i < D_MODEL; i += 256) {
        float v = xr[i];
        s += v * v;
    }
    red[threadIdx.x] = s;
    __syncthreads();
    for (int off = 128; off > 0; off >>= 1) {
        if (threadIdx.x < off) red[threadIdx.x] += red[threadIdx.x + off];
        __syncthreads();
    }
    const float scale = rsqrtf(red[0] * (1.f / D_MODEL) + EPS_F);
    float* xo = xn + (size_t)row * D_MODEL;
    for (int i = threadIdx.x; i < D_MODEL; i += 256)
        xo[i] = xr[i] * scale * normw[i];
}

// ---------------------------------------------------------------------------
// WMMA fp32 GEMM: C[M,N] = A[M,K] @ B[K,N] (+ residual).
// Block tile 64x128, BK=32, 256 threads = 8 waves, wave tile 32x32 (2x2 WMMA).
// Double-buffered LDS staging via GLOBAL_LOAD_ASYNC_TO_LDS_B128 when available.
// ---------------------------------------------------------------------------
template<bool ADD_RES>
__global__ __launch_bounds__(256)
void wmma_gemm_f32_kernel(const float* __restrict__ A, const float* __restrict__ B,
                          float* __restrict__ C, const float* __restrict__ res,
                          int M, int N, int K) {
    constexpr int BM = 64, BN = 128, BK = 32, PAD = 4;
#if HAVE_ASYNC_LDS
    __shared__ __align__(16) float As[2][BM][BK + PAD];   // row stride 144B (16B mult.)
    __shared__ __align__(16) float Bs[2][BK][BN + PAD];   // row stride 528B (16B mult.)
#else
    __shared__ __align__(16) float As[1][BM][BK + PAD];
    __shared__ __align__(16) float Bs[1][BK][BN + PAD];
#endif

    const int tid   = threadIdx.x;
    const int lane  = tid & 31;
    const int wave  = tid >> 5;
    const int wm    = (wave >> 2) * 32;   // wave M offset in block tile (0/32)
    const int wn    = (wave & 3) * 32;    // wave N offset (0..96)
    const int lmod  = lane & 15;
    const int lhalf = lane >> 4;
    const int blockM = blockIdx.y * BM;
    const int blockN = blockIdx.x * BN;

    v8f acc[2][2];
    #pragma unroll
    for (int i = 0; i < 2; i++)
        #pragma unroll
        for (int j = 0; j < 2; j++)
            #pragma unroll
            for (int e = 0; e < 8; e++) acc[i][j][e] = 0.f;

#if HAVE_ASYNC_LDS
    // Issue async copies of the (A,B) slab at k0 into LDS buffer `buf`.
    auto issue_tiles = [&](int k0, int buf) {
        // A tile: 64x32 floats = 512 16B quads, 2 per thread
        #pragma unroll
        for (int t = 0; t < 2; t++) {
            int q = tid + t * 256;
            int r = q >> 3, cq = (q & 7) * 4;
            async_cp16(&A[(size_t)(blockM + r) * K + (k0 + cq)], &As[buf][r][cq]);
        }
        // B tile: 32x128 floats = 1024 quads, 4 per thread
        #pragma unroll
        for (int t = 0; t < 4; t++) {
            int q = tid + t * 256;
            int r = q >> 5, cq = (q & 31) * 4;
            async_cp16(&B[(size_t)(k0 + r) * N + (blockN + cq)], &Bs[buf][r][cq]);
        }
    };
#endif

    const int nk = K / BK;
#if HAVE_ASYNC_LDS
    issue_tiles(0, 0);
#endif
    for (int ki = 0; ki < nk; ++ki) {
#if HAVE_ASYNC_LDS
        const int buf = ki & 1;
        wait_async0();          // this wave's copies for `buf` are done
        __syncthreads();        // all waves done copying `buf` AND done reading buf^1
        if (ki + 1 < nk) issue_tiles((ki + 1) * BK, buf ^ 1);  // overlaps compute
#else
        const int buf = 0;
        const int k0 = ki * BK;
        for (int i = tid; i < BM * BK; i += 256) {
            int r = i >> 5, c = i & 31;
            As[0][r][c] = A[(size_t)(blockM + r) * K + (k0 + c)];
        }
        for (int i = tid; i < BK * BN; i += 256) {
            int r = i >> 7, c = i & 127;
            Bs[0][r][c] = B[(size_t)(k0 + r) * N + (blockN + c)];
        }
        __syncthreads();
#endif

        #pragma unroll
        for (int kk = 0; kk < BK; kk += 4) {
            v2f afr[2], bfr[2];
            #pragma unroll
            for (int tm = 0; tm < 2; tm++) {
                // A 16x4: lanes 0-15 -> K = kk,kk+1 ; lanes 16-31 -> K = kk+2,kk+3
                afr[tm].x = As[buf][wm + tm * 16 + lmod][kk + lhalf * 2];
                afr[tm].y = As[buf][wm + tm * 16 + lmod][kk + lhalf * 2 + 1];
            }
            #pragma unroll
            for (int tn = 0; tn < 2; tn++) {
                // B 4x16: row striped across lanes, halves mirror A's K split
                bfr[tn].x = Bs[buf][kk + lhalf * 2][wn + tn * 16 + lmod];
                bfr[tn].y = Bs[buf][kk + lhalf * 2 + 1][wn + tn * 16 + lmod];
            }
            #pragma unroll
            for (int tm = 0; tm < 2; tm++)
                #pragma unroll
                for (int tn = 0; tn < 2; tn++)
                    acc[tm][tn] = __builtin_amdgcn_wmma_f32_16x16x4_f32(
                        false, afr[tm], false, bfr[tn],
                        (short)0, acc[tm][tn], false, false);
        }
#if !HAVE_ASYNC_LDS
        __syncthreads();
#endif
    }

    // C/D layout: lane<16: VGPR r -> M=r, N=lane ; lane>=16: VGPR r -> M=8+r, N=lane-16
    #pragma unroll
    for (int tm = 0; tm < 2; tm++)
        #pragma unroll
        for (int tn = 0; tn < 2; tn++)
            #pragma unroll
            for (int r = 0; r < 8; r++) {
                int row = blockM + wm + tm * 16 + lhalf * 8 + r;
                int col = blockN + wn + tn * 16 + lmod;
                float v = acc[tm][tn][r];
                if (ADD_RES) v += res[(size_t)row * N + col];
                C[(size_t)row * N + col] = v;
            }
}

// ---------------------------------------------------------------------------
// Kernel 3: causal depthwise conv (width 4) + bias + SiLU
// ---------------------------------------------------------------------------
__global__ __launch_bounds__(256)
void conv_silu_kernel(const float* __restrict__ xz,
                      const float* __restrict__ cw,
                      const float* __restrict__ cb,
                      float* __restrict__ xconv) {
    size_t idx = (size_t)blockIdx.x * 256 + threadIdx.x;  // over BATCH*SEQ*D_INNER
    int d = (int)(idx & (D_INNER - 1));
    int s = (int)((idx >> 11) & (SEQ - 1));
    int b = (int)(idx >> 21);
    float acc = cb[d];
    #pragma unroll
    for (int k = 0; k < D_CONV; k++) {
        int sp = s - (D_CONV - 1) + k;
        if (sp >= 0)
            acc += xz[(((size_t)b * SEQ + sp) * (2 * D_INNER)) + d] * cw[d * D_CONV + k];
    }
    xconv[idx] = acc / (1.f + __expf(-acc));  // SiLU
}

// ---------------------------------------------------------------------------
// Kernel 4: ssm = x_conv @ W_xproj   (K=2048, N=33)  one block per (b,s) row
// ---------------------------------------------------------------------------
__global__ __launch_bounds__(256)
void xproj_kernel(const float* __restrict__ xconv,
                  const float* __restrict__ Wx,
                  float* __restrict__ ssm) {
    __shared__ float sh[D_INNER];
    const int row = blockIdx.x;
    for (int i = threadIdx.x; i < D_INNER; i += 256)
        sh[i] = xconv[(size_t)row * D_INNER + i];
    __syncthreads();
    const int wave = threadIdx.x >> 5;
    const int lane = threadIdx.x & 31;
    for (int n = wave; n < 2 * D_STATE + 1; n += 8) {
        float p = 0.f;
        for (int k = lane; k < D_INNER; k += 32)
            p += sh[k] * Wx[k * (2 * D_STATE + 1) + n];
        for (int off = 16; off > 0; off >>= 1)
            p += __shfl_down(p, off, 32);
        if (lane == 0) ssm[(size_t)row * (2 * D_STATE + 1) + n] = p;
    }
}

// ---------------------------------------------------------------------------
// Kernel 5: selective scan; one thread per (b,d) channel, 16 states in regs.
// Writes gated y back into xconv in place (same-thread read-then-write).
// ---------------------------------------------------------------------------
__global__ __launch_bounds__(256)
void scan_kernel(const float* __restrict__ ssm,
                 float* __restrict__ xconv,
                 const float* __restrict__ xz,        // z = xz[..., D_INNER:]
                 const float* __restrict__ W_dt,
                 const float* __restrict__ b_dt,
                 const float* __restrict__ A_log,
                 const float* __restrict__ Dp) {
    const int t = blockIdx.x * 256 + threadIdx.x;     // 0..4095
    const int b = t >> 11;
    const int d = t & (D_INNER - 1);
    const float wdt = W_dt[d];
    const float bdt = b_dt[d];
    const float Dd  = Dp[d];
    float Ac[D_STATE];
    #pragma unroll
    for (int n = 0; n < D_STATE; n++) Ac[n] = -__expf(A_log[d * D_STATE + n]);
    float h[D_STATE];
    #pragma unroll
    for (int n = 0; n < D_STATE; n++) h[n] = 0.f;

    for (int s = 0; s < SEQ; s++) {
        const int row = b * SEQ + s;
        const float* sr = ssm + (size_t)row * (2 * D_STATE + 1);
        float u  = sr[0] * wdt + bdt;
        float dt = (u > 20.f) ? u : log1pf(__expf(u));   // softplus
        float xc = xconv[(size_t)row * D_INNER + d];
        float zv = xz[(size_t)row * (2 * D_INNER) + D_INNER + d];
        float dx = dt * xc;
        float y = 0.f;
        #pragma unroll
        for (int n = 0; n < D_STATE; n++) {
            float Ab = __expf(dt * Ac[n]);
            h[n] = Ab * h[n] + dx * sr[1 + n];
            y += h[n] * sr[1 + D_STATE + n];
        }
        y += xc * Dd;
        y *= zv / (1.f + __expf(-zv));                   // gate with silu(z)
        xconv[(size_t)row * D_INNER + d] = y;
    }
}

// ---------------------------------------------------------------------------
extern "C" void kernel_launch(void* const* d_in, const int* in_sizes, int n_in,
                              void* d_out, int out_size, void* d_ws, size_t ws_size,
                              hipStream_t stream) {
    const float* x      = (const float*)d_in[0];
    const float* W_in   = (const float*)d_in[1];
    const float* conv_w = (const float*)d_in[2];
    const float* conv_b = (const float*)d_in[3];
    const float* W_xprj = (const float*)d_in[4];
    const float* W_dt   = (const float*)d_in[5];
    const float* b_dt   = (const float*)d_in[6];
    const float* A_log  = (const float*)d_in[7];
    const float* Dp     = (const float*)d_in[8];
    const float* W_out  = (const float*)d_in[9];
    const float* norm_w = (const float*)d_in[10];
    float* out = (float*)d_out;

    float* ws    = (float*)d_ws;
    float* xn    = ws;                                         // 2048*1024
    float* xz    = xn + (size_t)NROWS * D_MODEL;               // 2048*4096
    float* xconv = xz + (size_t)NROWS * (2 * D_INNER);         // 2048*2048
    float* ssm   = xconv + (size_t)NROWS * D_INNER;            // 2048*33

    // 1) rmsnorm
    rmsnorm_kernel<<<NROWS, 256, 0, stream>>>(x, norm_w, xn);

    // 2) xz = xn @ W_in   (M=2048, N=4096, K=1024)
    wmma_gemm_f32_kernel<false><<<dim3(4096 / 128, NROWS / 64), 256, 0, stream>>>(
        xn, W_in, xz, nullptr, NROWS, 2 * D_INNER, D_MODEL);

    // 3) depthwise causal conv + silu
    conv_silu_kernel<<<(NROWS * D_INNER) / 256, 256, 0, stream>>>(xz, conv_w, conv_b, xconv);

    // 4) ssm = x_conv @ W_xproj
    xproj_kernel<<<NROWS, 256, 0, stream>>>(xconv, W_xprj, ssm);

    // 5) selective scan + gate (xconv becomes y2 in place)
    scan_kernel<<<(BATCH * D_INNER) / 256, 256, 0, stream>>>(
        ssm, xconv, xz, W_dt, b_dt, A_log, Dp);

    // 6) out = x + y2 @ W_out   (M=2048, N=1024, K=2048)
    wmma_gemm_f32_kernel<true><<<dim3(1024 / 128, NROWS / 64), 256, 0, stream>>>(
        xconv, W_out, out, x, NROWS, D_MODEL, D_INNER);
}